// LSTM_24970939859454
// MI455X (gfx1250) — compile-verified
//
#include <hip/hip_runtime.h>

typedef __attribute__((ext_vector_type(2))) float v2f;
typedef __attribute__((ext_vector_type(8))) float v8f;

constexpr int V_ = 10000, E_ = 1024, H_ = 1024, B_ = 32, S_ = 512;
constexpr int G4 = 4 * H_;        // 4096 gate columns
#define GBLK 16                   // persistent-kernel grid (co-resident for sure)

__device__ __forceinline__ float fsig(float x) { return 1.0f / (1.0f + __expf(-x)); }
__device__ __forceinline__ float ftanh(float x) { return fmaf(2.0f, fsig(2.0f * x), -1.0f); }

__device__ __forceinline__ v8f wmma_f32(v2f a, v2f b, v8f c) {
  // D = A(16x4 f32) * B(4x16 f32) + C, full fp32
  return __builtin_amdgcn_wmma_f32_16x16x4_f32(false, a, false, b, (short)0, c, false, false);
}

// ---------------------------------------------------------------------------
// Kernel A: fused embedding gather + input GEMM.
// xg[s*B + b, g] = dot(emb[x[b,s]], Wx[g,:]) + bx[g] + bh[g]
// Register-blocked: one wave = 32(M) x 64(N) output block
//   (2 M-tiles x 4 N-tiles, 8 accumulators; 6 loads feed 8 WMMAs per k-step).
// ---------------------------------------------------------------------------
__global__ __launch_bounds__(256) void k_embed_gemm(
    const int* __restrict__ x, const float* __restrict__ emb,
    const float* __restrict__ Wx, const float* __restrict__ bx,
    const float* __restrict__ bh, float* __restrict__ xg) {
  const int lane = threadIdx.x & 31;
  const int wib  = threadIdx.x >> 5;
  const int grp  = blockIdx.x * 8 + wib;      // 0..32767
  const int nGrpN = (G4 / 16) / 4;            // 64 groups of 4 N-tiles
  const int m0 = (grp / nGrpN) * 32;          // 512 M-groups of 32 rows
  const int n0 = (grp % nGrpN) * 64;
  const int lm = lane & 15, half = lane >> 4, koff = half * 2;

  // Two A rows per lane (rows m0+lm and m0+16+lm); row ordered as s*32 + b.
  const float* arow[2];
#pragma unroll
  for (int mi = 0; mi < 2; ++mi) {
    const int m = m0 + 16 * mi + lm;
    const int b = m & 31, s = m >> 5;
    arow[mi] = emb + (size_t)x[b * S_ + s] * E_;
  }
  const float* brow[4];
#pragma unroll
  for (int q = 0; q < 4; ++q)
    brow[q] = Wx + (size_t)(n0 + 16 * q + lm) * E_;

  v8f acc[2][4] = {};
  for (int k = 0; k < E_; k += 4) {
    v2f a0 = *(const v2f*)(arow[0] + k + koff);
    v2f a1 = *(const v2f*)(arow[1] + k + koff);
#pragma unroll
    for (int q = 0; q < 4; ++q) {
      v2f bb = *(const v2f*)(brow[q] + k + koff);
      acc[0][q] = wmma_f32(a0, bb, acc[0][q]);
      acc[1][q] = wmma_f32(a1, bb, acc[1][q]);
    }
  }
#pragma unroll
  for (int q = 0; q < 4; ++q) {
    const int col = n0 + 16 * q + lm;
    const float bias = bx[col] + bh[col];
#pragma unroll
    for (int mi = 0; mi < 2; ++mi)
#pragma unroll
      for (int r = 0; r < 8; ++r) {
        const int row = m0 + 16 * mi + r + 8 * half;
        xg[(size_t)row * G4 + col] = acc[mi][q][r] + bias;
      }
  }
}

// ---------------------------------------------------------------------------
// Device-wide barrier: monotone ticket, phase counted per call.
// ---------------------------------------------------------------------------
__device__ __forceinline__ void gbar(unsigned* ticket, unsigned phase) {
  __threadfence();          // release h/c stores to device scope
  __syncthreads();
  if (threadIdx.x == 0) {
    atomicAdd(ticket, 1u);
    while (__hip_atomic_load(ticket, __ATOMIC_RELAXED, __HIP_MEMORY_SCOPE_AGENT) <
           GBLK * phase) {
      __builtin_amdgcn_s_sleep(1);
    }
  }
  __syncthreads();
  __threadfence();          // acquire other blocks' h stores
}

// ---------------------------------------------------------------------------
// Kernel R: persistent LSTM recurrence. 16 blocks x 8 waves = 128 waves.
// Each wave owns a 16(batch) x 16(hidden-col) tile and computes all four
// gate blocks (f,i,g,o): A fragment amortized over 4 accumulators.
// Wh (16 MB) stays hot in the 192 MB L2 across all 512 steps.
// h double-buffered across steps; c is wave-private.
// ---------------------------------------------------------------------------
__global__ __launch_bounds__(256) void k_lstm(
    const float* __restrict__ xg, const float* __restrict__ Wh,
    float* __restrict__ hbuf, float* __restrict__ cbuf,
    float* __restrict__ outs, unsigned* __restrict__ ticket) {
  const int lane = threadIdx.x & 31;
  const int wid  = blockIdx.x * 8 + (threadIdx.x >> 5);   // 0..127
  const int m0 = (wid >> 6) * 16;                         // batch tile (0 or 16)
  const int j0 = (wid & 63) * 16;                         // hidden-col tile
  const int lm = lane & 15, half = lane >> 4, koff = half * 2;
  const int col = j0 + lm;

  // zero-init this wave's c tile and h tile (buffer 0)
#pragma unroll
  for (int r = 0; r < 8; ++r) {
    const int row = m0 + r + 8 * half;
    cbuf[row * H_ + col] = 0.0f;
    hbuf[row * H_ + col] = 0.0f;
  }
  unsigned phase = 1;
  gbar(ticket, phase++);

  const float* wF = Wh + (size_t)(0 * H_ + j0 + lm) * H_;
  const float* wI = Wh + (size_t)(1 * H_ + j0 + lm) * H_;
  const float* wG = Wh + (size_t)(2 * H_ + j0 + lm) * H_;
  const float* wO = Wh + (size_t)(3 * H_ + j0 + lm) * H_;

  for (int t = 0; t < S_; ++t) {
    const float* hprev = hbuf + (size_t)(t & 1) * (B_ * H_);
    float*       hnext = hbuf + (size_t)((t + 1) & 1) * (B_ * H_);
    const float* arow  = hprev + (m0 + lm) * H_;

    v8f aF = {}, aI = {}, aG = {}, aO = {};
    for (int k = 0; k < H_; k += 4) {
      v2f a = *(const v2f*)(arow + k + koff);
      aF = wmma_f32(a, *(const v2f*)(wF + k + koff), aF);
      aI = wmma_f32(a, *(const v2f*)(wI + k + koff), aI);
      aG = wmma_f32(a, *(const v2f*)(wG + k + koff), aG);
      aO = wmma_f32(a, *(const v2f*)(wO + k + koff), aO);
    }

#pragma unroll
    for (int r = 0; r < 8; ++r) {
      const int row = m0 + r + 8 * half;                  // batch index
      const size_t xb = ((size_t)t * B_ + row) * G4 + col;
      float f = fsig (aF[r] + xg[xb + 0 * H_]);
      float i = fsig (aI[r] + xg[xb + 1 * H_]);
      float g = ftanh(aG[r] + xg[xb + 2 * H_]);
      float o = fsig (aO[r] + xg[xb + 3 * H_]);
      float c = f * cbuf[row * H_ + col] + i * g;
      cbuf[row * H_ + col] = c;
      float h = o * ftanh(c);
      hnext[row * H_ + col] = h;
      outs[((size_t)row * S_ + t) * H_ + col] = h;        // outs[B,S,H]
    }
    gbar(ticket, phase++);
  }
}

// ---------------------------------------------------------------------------
// Kernel C: decoder GEMM. out[m, v] = dot(outs[m,:], Wd[v,:]) + bd[v]
// M=16384, N=10000 = 125 groups of 5 exact 16-tiles, K=1024.
// Register-blocked: one wave = 32(M) x 80(N) output block
//   (2 M-tiles x 5 N-tiles, 10 accumulators; 7 loads feed 10 WMMAs).
// ---------------------------------------------------------------------------
__global__ __launch_bounds__(256) void k_decoder(
    const float* __restrict__ outs, const float* __restrict__ Wd,
    const float* __restrict__ bd, float* __restrict__ out) {
  const int lane = threadIdx.x & 31;
  const int grp  = blockIdx.x * 8 + (threadIdx.x >> 5);   // 0..63999
  const int nGrpN = (V_ / 16) / 5;            // 125 groups of 5 N-tiles
  const int m0 = (grp / nGrpN) * 32;          // 512 M-groups of 32 rows
  const int n0 = (grp % nGrpN) * 80;
  const int lm = lane & 15, half = lane >> 4, koff = half * 2;

  const float* arow[2];
#pragma unroll
  for (int mi = 0; mi < 2; ++mi)
    arow[mi] = outs + (size_t)(m0 + 16 * mi + lm) * H_;
  const float* brow[5];
#pragma unroll
  for (int q = 0; q < 5; ++q)
    brow[q] = Wd + (size_t)(n0 + 16 * q + lm) * H_;

  v8f acc[2][5] = {};
  for (int k = 0; k < H_; k += 4) {
    v2f a0 = *(const v2f*)(arow[0] + k + koff);
    v2f a1 = *(const v2f*)(arow[1] + k + koff);
#pragma unroll
    for (int q = 0; q < 5; ++q) {
      v2f bb = *(const v2f*)(brow[q] + k + koff);
      acc[0][q] = wmma_f32(a0, bb, acc[0][q]);
      acc[1][q] = wmma_f32(a1, bb, acc[1][q]);
    }
  }
#pragma unroll
  for (int q = 0; q < 5; ++q) {
    const int col = n0 + 16 * q + lm;
    const float bias = bd[col];
#pragma unroll
    for (int mi = 0; mi < 2; ++mi)
#pragma unroll
      for (int r = 0; r < 8; ++r) {
        const int row = m0 + 16 * mi + r + 8 * half;
        out[(size_t)row * V_ + col] = acc[mi][q][r] + bias;
      }
  }
}

extern "C" void kernel_launch(void* const* d_in, const int* in_sizes, int n_in,
                              void* d_out, int out_size, void* d_ws, size_t ws_size,
                              hipStream_t stream) {
  (void)in_sizes; (void)n_in; (void)out_size; (void)ws_size;
  const int*   x   = (const int*)d_in[0];
  const float* emb = (const float*)d_in[1];
  const float* Wx  = (const float*)d_in[2];
  const float* bx  = (const float*)d_in[3];
  const float* Wh  = (const float*)d_in[4];
  const float* bh  = (const float*)d_in[5];
  const float* Wd  = (const float*)d_in[6];
  const float* bd  = (const float*)d_in[7];
  float* out = (float*)d_out;

  // workspace layout (floats)
  float* xg    = (float*)d_ws;                         // [S*B, 4H]  = 256 MB
  float* outsb = xg + (size_t)S_ * B_ * G4;            // [B, S, H]  =  64 MB
  float* hb    = outsb + (size_t)B_ * S_ * H_;         // 2 x [B, H] double buffer
  float* cb    = hb + 2 * B_ * H_;                     // [B, H]
  unsigned* ticket = (unsigned*)(cb + B_ * H_);

  hipMemsetAsync(ticket, 0, sizeof(unsigned), stream); // graph-capturable

  // A: 512 M-groups * 64 N-groups = 32768 waves / 8 = 4096 blocks
  k_embed_gemm<<<4096, 256, 0, stream>>>(x, emb, Wx, bx, bh, xg);
  // R: persistent recurrence
  k_lstm<<<GBLK, 256, 0, stream>>>(xg, Wh, hb, cb, outsb, ticket);
  // C: 512 M-groups * 125 N-groups = 64000 waves / 8 = 8000 blocks
  k_decoder<<<8000, 256, 0, stream>>>(outsb, Wd, bd, out);
}